// SpectralNetTrainer_16621523436057
// MI455X (gfx1250) — compile-verified
//
#include <hip/hip_runtime.h>
#include <hip/hip_bf16.h>

typedef __attribute__((ext_vector_type(2))) float v2f;
typedef __attribute__((ext_vector_type(8))) float v8f;

#define N_PTS 4096
#define DIMS  512
#define KSEL  31          // n_nbg + 1

// ---------------------------------------------------------------------------
// Kernel 1: per-row squared norms. One wave (32 lanes) per row, float4 loads.
// ---------------------------------------------------------------------------
__global__ __launch_bounds__(256)
void spnet_rownorm(const float* __restrict__ X, float* __restrict__ sq,
                   int n, int d) {
  const int row  = blockIdx.x * 8 + (threadIdx.x >> 5);
  const int lane = threadIdx.x & 31;
  const float* xr = X + (size_t)row * d;
  float acc = 0.0f;
  for (int k = lane * 4; k < d; k += 32 * 4) {
    float4 v = *(const float4*)(xr + k);
    acc += v.x * v.x + v.y * v.y + v.z * v.z + v.w * v.w;
  }
  #pragma unroll
  for (int off = 16; off >= 1; off >>= 1)
    acc += __shfl_xor(acc, off, 32);
  if (lane == 0) sq[row] = acc;
}

// ---------------------------------------------------------------------------
// Kernel 2: WMMA fp32 GEMM -> d2 = max(sq_i + sq_j - 2*X.X^T, 0), in-place
// into d_out (used as scratch). 128x64 macro tile / 256 threads (8 waves),
// each wave computes a 16x64 strip with 4 f32 accumulators.
// ---------------------------------------------------------------------------
#define TILE_M 128
#define TILE_N 64
#define KC     32
#define LDA    36   // 36 floats = 144 B row stride: 16B-aligned float4 stores
#define LDB    36

__global__ __launch_bounds__(256)
void spnet_gram_d2(const float* __restrict__ X, const float* __restrict__ sq,
                   float* __restrict__ d2, int n, int d) {
  __shared__ float As[TILE_M * LDA];
  __shared__ float Bs[TILE_N * LDB];

  const int rowBase = blockIdx.y * TILE_M;
  const int colBase = blockIdx.x * TILE_N;
  const int t     = threadIdx.x;
  const int lane  = t & 31;
  const int wave  = t >> 5;        // 0..7
  const int lhalf = lane >> 4;     // 0 or 1 (K-pair select per ISA layout)
  const int l15   = lane & 15;

  v8f c[4];
  #pragma unroll
  for (int i = 0; i < 4; ++i) c[i] = (v8f){0,0,0,0,0,0,0,0};

  for (int kt = 0; kt < d; kt += KC) {
    // cooperative stage: A = 128x32 (1024 float4 / 256 thr = 4 each)
    #pragma unroll
    for (int i = 0; i < 4; ++i) {
      int idx4 = t + i * 256;
      int r    = idx4 >> 3;            // 8 float4 per row
      int cq   = (idx4 & 7) << 2;
      float4 v = *(const float4*)(X + (size_t)(rowBase + r) * d + kt + cq);
      *(float4*)(&As[r * LDA + cq]) = v;
    }
    // B = 64x32 (512 float4 / 256 thr = 2 each)
    #pragma unroll
    for (int i = 0; i < 2; ++i) {
      int idx4 = t + i * 256;
      int r    = idx4 >> 3;
      int cq   = (idx4 & 7) << 2;
      float4 v = *(const float4*)(X + (size_t)(colBase + r) * d + kt + cq);
      *(float4*)(&Bs[r * LDB + cq]) = v;
    }
    __syncthreads();

    const int ar = wave * 16 + l15;   // this lane's A row in the macro tile
    #pragma unroll
    for (int k0 = 0; k0 < KC; k0 += 4) {
      // A frag: lanes 0-15 hold K={k0,k0+1}, lanes 16-31 hold K={k0+2,k0+3}
      v2f a = *(const v2f*)(&As[ar * LDA + k0 + lhalf * 2]);
      #pragma unroll
      for (int nsub = 0; nsub < 4; ++nsub) {
        v2f b = *(const v2f*)(&Bs[(nsub * 16 + l15) * LDB + k0 + lhalf * 2]);
        c[nsub] = __builtin_amdgcn_wmma_f32_16x16x4_f32(
            false, a, false, b, (short)0, c[nsub], false, false);
      }
    }
    __syncthreads();
  }

  // epilogue: d2 = clamp(sq_i + sq_j - 2*acc, 0)
  const int rBase = rowBase + wave * 16 + lhalf * 8;
  #pragma unroll
  for (int nsub = 0; nsub < 4; ++nsub) {
    const int j   = colBase + nsub * 16 + l15;
    const float s = sq[j];
    #pragma unroll
    for (int r = 0; r < 8; ++r) {
      const int i = rBase + r;
      float v = fmaxf(sq[i] + s - 2.0f * c[nsub][r], 0.0f);
      d2[(size_t)i * n + j] = v;
    }
  }
}

// ---------------------------------------------------------------------------
// Kernel 3: exact k-th smallest per row via 32-pass radix select on float
// bit patterns (all d2 >= 0, so uint order == float order). Row staged in LDS.
// ---------------------------------------------------------------------------
__global__ __launch_bounds__(256)
void spnet_kth_select(const float* __restrict__ d2, float* __restrict__ thresh,
                      float* __restrict__ invDenom, int n, int k) {
  __shared__ unsigned srow[N_PTS];
  __shared__ int cnt;
  const int row = blockIdx.x;
  const float* dr = d2 + (size_t)row * n;

  for (int j = threadIdx.x; j < n; j += blockDim.x)
    srow[j] = __float_as_uint(dr[j]);
  __syncthreads();

  unsigned prefix = 0;
  int kk = k;                                // 1-based rank wanted
  for (int bit = 31; bit >= 0; --bit) {
    if (threadIdx.x == 0) cnt = 0;
    __syncthreads();
    const unsigned hiMask = (bit == 31) ? 0u : (0xFFFFFFFFu << (bit + 1));
    int local = 0;
    for (int j = threadIdx.x; j < n; j += blockDim.x) {
      unsigned u = srow[j];
      local += (((u & hiMask) == prefix) && (((u >> bit) & 1u) == 0u)) ? 1 : 0;
    }
    atomicAdd(&cnt, local);
    __syncthreads();
    const int c0 = cnt;
    if (kk > c0) { kk -= c0; prefix |= (1u << bit); }
    __syncthreads();
  }

  if (threadIdx.x == 0) {
    const float tval = __uint_as_float(prefix);    // k-th smallest d2
    thresh[row] = tval;
    const float dis = sqrtf(fmaxf(tval, 1e-12f));  // local scale (max kNN dist)
    const float s   = fmaxf(dis, 1e-8f);
    invDenom[row]   = 1.0f / (s * s);
  }
}

// ---------------------------------------------------------------------------
// Kernel 4: gated Gaussian kernel + symmetrization, in place over d_out.
// One WG per block pair (bi<=bj): loads tiles d2[i,j] and d2[j,i] into LDS,
// computes v = 0.5*(W_ij + W_ji) once, writes both orientations coalesced.
// ---------------------------------------------------------------------------
#define ST 64
__global__ __launch_bounds__(256)
void spnet_sym_w(float* __restrict__ m, const float* __restrict__ thresh,
                 const float* __restrict__ invDenom, int n) {
  const int bi = blockIdx.y, bj = blockIdx.x;
  if (bi > bj) return;
  __shared__ float Ta[ST][ST + 1];
  __shared__ float Tb[ST][ST + 1];
  const int t   = threadIdx.x;
  const int tc  = t & 63;
  const int tr0 = t >> 6;       // 0..3
  const size_t I0 = (size_t)bi * ST, J0 = (size_t)bj * ST;

  for (int r = tr0; r < ST; r += 4) {
    Ta[r][tc] = m[(I0 + r) * n + J0 + tc];
    Tb[r][tc] = m[(J0 + r) * n + I0 + tc];
  }
  __syncthreads();

  for (int r = tr0; r < ST; r += 4) {
    const int i = (int)I0 + r;
    const int j = (int)J0 + tc;
    const float ti = thresh[i], di = invDenom[i];
    const float a = Ta[r][tc];         // d2[i][j]
    const float b = Tb[tc][r];         // d2[j][i]
    const float wa = (a <= ti)        ? expf(-a * di)          : 0.0f;
    const float wb = (b <= thresh[j]) ? expf(-b * invDenom[j]) : 0.0f;
    const float v = 0.5f * (wa + wb);
    m[(size_t)i * n + j] = v;
    Ta[r][tc] = v;                     // stash for transposed write
  }
  __syncthreads();

  if (bi != bj) {
    for (int r = tr0; r < ST; r += 4)
      m[(J0 + r) * n + I0 + tc] = Ta[tc][r];
  }
}

// ---------------------------------------------------------------------------
extern "C" void kernel_launch(void* const* d_in, const int* in_sizes, int n_in,
                              void* d_out, int out_size, void* d_ws, size_t ws_size,
                              hipStream_t stream) {
  const int n = N_PTS, d = DIMS;          // matches reference setup_inputs()
  const float* X = (const float*)d_in[0];
  float* out = (float*)d_out;             // also used as d2 scratch (in place)
  float* sq     = (float*)d_ws;           // [n]
  float* thresh = sq + n;                 // [n]
  float* invDen = thresh + n;             // [n]

  spnet_rownorm<<<n / 8, 256, 0, stream>>>(X, sq, n, d);

  dim3 g2(n / TILE_N, n / TILE_M);
  spnet_gram_d2<<<g2, 256, 0, stream>>>(X, sq, out, n, d);

  spnet_kth_select<<<n, 256, 0, stream>>>(out, thresh, invDen, n, KSEL);

  dim3 g4(n / ST, n / ST);
  spnet_sym_w<<<g4, 256, 0, stream>>>(out, thresh, invDen, n);
}